// GraphAttnBias_29652454212053
// MI455X (gfx1250) — compile-verified
//
#include <hip/hip_runtime.h>

typedef float v2f __attribute__((ext_vector_type(2)));
typedef float v8f __attribute__((ext_vector_type(8)));

#define H_      32
#define N_      64
#define B_      64
#define NP1     65
#define MHD_    5
#define KDIM    160        // MHD_*H_
#define ASTRIDE 164        // padded A row stride (floats): conflict-free column reads
#define WAVES   4

// One wave computes a tile of 16 (n,m)-pairs: E(16x32) = BAG(16x160) @ Wd(160x32)
// using V_WMMA_F32_16X16X4_F32, then fuses attn_bias + spatial + E/sp into out.
__global__ __launch_bounds__(128)
void edge_bias_kernel(const float* __restrict__ attn_bias,
                      const int*   __restrict__ spatial_pos,
                      const int*   __restrict__ edge_input,
                      const float* __restrict__ W_edge,
                      const float* __restrict__ W_dist,
                      const float* __restrict__ W_spatial,
                      float* __restrict__ out)
{
    __shared__ float wd_lds[KDIM * H_];            // 20 KB: Wd row-major [160][32]
    __shared__ float a_lds[WAVES * 16 * ASTRIDE];  // 41 KB: per-wave A tiles

    const int lane = threadIdx.x & 31;
    const int wv   = threadIdx.x >> 5;

    // stage Wd = W_dist.reshape(-1,H,H)[:5] == first 5120 floats, K-major
    for (int i = threadIdx.x; i < KDIM * H_; i += 128) wd_lds[i] = W_dist[i];

    const int tile = blockIdx.x * WAVES + wv;      // 16384 tiles total, exact
    const int bi   = tile >> 8;
    const int rem  = tile & 255;
    const int n    = rem >> 2;
    const int m0   = (rem & 3) << 4;

    // warm the attn_bias lines this tile touches (one 64B line per head)
    {
        const int o = (bi * H_ + lane) * (NP1 * NP1) + (n + 1) * NP1 + (m0 + 1);
        __builtin_prefetch(&attn_bias[o], 0, 0);
    }

    __syncthreads();  // wd_lds ready

    float* A = a_lds + wv * (16 * ASTRIDE);

    // ---- build A = embedding-bag tile (16 pairs x 160); lane indexes head h ----
    const int* ein = edge_input + ((bi * N_ + n) * N_ + m0) * (MHD_ * 3);
    for (int p = 0; p < 16; ++p) {
        const int* ep = ein + p * (MHD_ * 3);
#pragma unroll
        for (int d = 0; d < MHD_; ++d) {
            const int i0 = ep[d * 3 + 0];
            const int i1 = ep[d * 3 + 1];
            const int i2 = ep[d * 3 + 2];
            float s = 0.0f; int c = 0;
            if (i0) { s += W_edge[i0 * H_ + lane]; ++c; }   // padding_idx=0 excluded
            if (i1) { s += W_edge[i1 * H_ + lane]; ++c; }
            if (i2) { s += W_edge[i2 * H_ + lane]; ++c; }
            const float rc = (c == 3) ? 0x1.555556p-2f : ((c == 2) ? 0.5f : 1.0f);
            A[p * ASTRIDE + d * H_ + lane] = s * rc;        // c==0 -> s==0 -> 0
        }
    }
    // same-wave LDS ops are in order: A stores visible to the ds reads below

    // ---- E(16x32) = A(16x160) @ Wd(160x32): 40 K-steps x 2 N-halves ----
    // f32 16x16x4 fragment layout (ISA 7.12.2): lanes 0-15 hold K=0/1 in v0/v1,
    // lanes 16-31 hold K=2/3; B mirrors with N striped across lanes.
    v8f acc0 = {}; v8f acc1 = {};
    const int rowA  = lane & 15;
    const int koffA = (lane >> 4) << 1;  // 0 or 2
    const int colB  = lane & 15;
    for (int kk = 0; kk < KDIM / 4; ++kk) {
        const int k0 = kk * 4 + koffA;
        v2f a, b0, b1;
        a.x  = A[rowA * ASTRIDE + k0];
        a.y  = A[rowA * ASTRIDE + k0 + 1];
        b0.x = wd_lds[k0 * H_ + colB];
        b0.y = wd_lds[(k0 + 1) * H_ + colB];
        b1.x = wd_lds[k0 * H_ + 16 + colB];
        b1.y = wd_lds[(k0 + 1) * H_ + 16 + colB];
        acc0 = __builtin_amdgcn_wmma_f32_16x16x4_f32(false, a, false, b0,
                                                     (short)0, acc0, false, false);
        acc1 = __builtin_amdgcn_wmma_f32_16x16x4_f32(false, a, false, b1,
                                                     (short)0, acc1, false, false);
    }

    // ---- spill E to LDS (reuse this wave's A region; stride 33 => conflict-free reads) ----
    float* E = A;
    const int rsel = lane >> 4;
    const int nlo  = lane & 15;
#pragma unroll
    for (int j = 0; j < 8; ++j) {       // D layout: VGPR j = rows j / j+8
        E[(j + rsel * 8) * 33 + nlo]      = acc0[j];
        E[(j + rsel * 8) * 33 + 16 + nlo] = acc1[j];
    }

    // ---- fused epilogue: out = attn_bias + W_spatial[sp] + E * (1/clip(sp-1,1,5)) ----
    const int p  = lane & 15;           // pair within tile
    const int hb = lane >> 4;           // head half
    const int m  = m0 + p;
    const int sp = spatial_pos[(bi * N_ + n) * N_ + m];
    int spc = sp - 1; spc = spc < 1 ? 1 : (spc > MHD_ ? MHD_ : spc);
    const float inv = 1.0f / (float)spc;
    const float* Wsp = W_spatial + sp * H_;
    const int obase = bi * (H_ * NP1 * NP1) + (n + 1) * NP1 + (m + 1);
#pragma unroll
    for (int hh = 0; hh < 16; ++hh) {
        const int h = hb * 16 + hh;     // two 64B-contiguous half-wave segments
        const int o = obase + h * (NP1 * NP1);
        out[o] = attn_bias[o] + Wsp[h] + E[p * 33 + h] * inv;
    }
}

// Row 0 (65 cols) and column 0 (rows 1..64): out = attn_bias + w_token[h]
__global__ __launch_bounds__(256)
void border_kernel(const float* __restrict__ ab,
                   const float* __restrict__ w_token,
                   float* __restrict__ out)
{
    const int i = blockIdx.x * blockDim.x + threadIdx.x;
    if (i >= B_ * H_ * 129) return;
    const int e  = i % 129;
    const int bh = i / 129;
    const int h  = bh & (H_ - 1);
    const int r  = (e < NP1) ? 0 : (e - (NP1 - 1));
    const int c  = (e < NP1) ? e : 0;
    const int o  = bh * (NP1 * NP1) + r * NP1 + c;
    out[o] = ab[o] + w_token[h];
}

extern "C" void kernel_launch(void* const* d_in, const int* in_sizes, int n_in,
                              void* d_out, int out_size, void* d_ws, size_t ws_size,
                              hipStream_t stream)
{
    const float* attn_bias   = (const float*)d_in[0];
    const int*   spatial_pos = (const int*)d_in[1];
    const int*   edge_input  = (const int*)d_in[2];
    // d_in[3] = attn_edge_type: unused by the reference computation
    const float* W_edge      = (const float*)d_in[4];
    const float* W_dist      = (const float*)d_in[5];
    const float* W_spatial   = (const float*)d_in[6];
    const float* w_token     = (const float*)d_in[7];
    float* out = (float*)d_out;

    const int tiles = B_ * N_ * (N_ / 16);  // 16384, divisible by WAVES
    edge_bias_kernel<<<tiles / WAVES, 128, 0, stream>>>(
        attn_bias, spatial_pos, edge_input, W_edge, W_dist, W_spatial, out);

    const int border_tot = B_ * H_ * 129;   // 264192
    border_kernel<<<(border_tot + 255) / 256, 256, 0, stream>>>(attn_bias, w_token, out);
}